// FlowLSTM_44555990728884
// MI455X (gfx1250) — compile-verified
//
#include <hip/hip_runtime.h>
#include <math.h>
#include <stdint.h>

typedef float v2f __attribute__((ext_vector_type(2)));
typedef float v8f __attribute__((ext_vector_type(8)));

constexpr int BATCH  = 4096;
constexpr int SEQ_T  = 19;
constexpr int IN_DIM = 17;
constexpr int HID    = 1024;

constexpr int BM = 64;   // batch rows per block tile
constexpr int BN = 64;   // hidden cols per block tile (per gate)
constexpr int KC = 32;   // K chunk staged in LDS
constexpr int KP = 36;   // padded LDS row stride (stride 36 mod 64 banks -> conflict-free frag reads)
constexpr int NCHUNK = HID / KC;   // 32

__device__ __forceinline__ float sigmoidf_(float x) {
    return 1.0f / (1.0f + __expf(-x));
}

// low 32 bits of a generic pointer into __shared__ == raw LDS byte offset (aperture mapping)
__device__ __forceinline__ uint32_t lds_off_(const void* p) {
    return (uint32_t)(uintptr_t)p;
}

// async global -> LDS, 16 bytes per lane, tracked by ASYNCcnt
__device__ __forceinline__ void async_b128_(uint32_t lds_byte_off, const void* gsrc) {
    uint64_t ga = (uint64_t)(uintptr_t)gsrc;
    asm volatile("global_load_async_to_lds_b128 %0, %1, off"
                 :: "v"(lds_byte_off), "v"(ga) : "memory");
}

// One LSTM timestep: gates = h_prev @ Whh^T + x_t @ Wih^T + b, then c/h update.
// grid = (HID/BN, BATCH/BM) = (16, 64), block = 256 (8 waves).
__global__ __launch_bounds__(256)
void lstm_step_kernel(const float* __restrict__ x,       // [B,T,IN]
                      const float* __restrict__ Wih,     // [4H,IN]
                      const float* __restrict__ Whh,     // [4H,H]
                      const float* __restrict__ bih,     // [4H]
                      const float* __restrict__ bhh,     // [4H]
                      const float* __restrict__ h_prev,  // [B,H]
                      float* __restrict__ h_next,        // [B,H]
                      float* __restrict__ c_state,       // [B,H] in/out
                      int t)
{
    __shared__ float sA[2][BM * KP];        // double-buffered h tile      (2 x 9216 B)
    __shared__ float sW[2][4 * BN * KP];    // double-buffered gate weights (2 x 36864 B)
    __shared__ float sX[BM * IN_DIM];       // x tile

    const int tid  = threadIdx.x;
    const int lane = tid & 31;
    const int wid  = tid >> 5;               // wave 0..7
    const int mq   = wid >> 2;               // 0..1  : which 32-row half of the M tile
    const int nsub = wid & 3;                // 0..3  : N subtile
    const int half = lane >> 4;              // K-half selector per ISA A/B frag layout
    const int l16  = lane & 15;

    const int b0 = blockIdx.y * BM;
    const int n0 = blockIdx.x * BN;

    // stage x tile [BM x IN_DIM] (plain stores; covered by first barrier)
    for (int lin = tid; lin < BM * IN_DIM; lin += 256) {
        int r = lin / IN_DIM, k = lin % IN_DIM;
        sX[r * IN_DIM + k] = x[((size_t)(b0 + r) * SEQ_T + t) * IN_DIM + k];
    }

    // issue the 10 async b128 loads (2 for A, 8 for W) for chunk `kt` into buffer `buf`
    auto issue_chunk = [&](int buf, int kt) {
        #pragma unroll
        for (int jj = 0; jj < 2; ++jj) {                 // A: 64 rows x 32 floats = 512 b128
            int lin = tid + jj * 256;
            int c4  = lin & 7;                           // 8 x float4 per row
            int r   = lin >> 3;                          // 0..63
            const float* src = h_prev + (size_t)(b0 + r) * HID + kt + c4 * 4;
            async_b128_(lds_off_(&sA[buf][r * KP + c4 * 4]), src);
        }
        #pragma unroll
        for (int jj = 0; jj < 8; ++jj) {                 // W: 4*64 rows x 32 floats = 2048 b128
            int lin = tid + jj * 256;
            int c4  = lin & 7;
            int r   = (lin >> 3) & 63;
            int g   = lin >> 9;                          // 0..3
            const float* src = Whh + (size_t)(g * HID + n0 + r) * HID + kt + c4 * 4;
            async_b128_(lds_off_(&sW[buf][(g * BN + r) * KP + c4 * 4]), src);
        }
    };

    v8f acc[2][4] = {};                      // 2 M-subtiles x 4 gates, 16x16 f32 tiles

    const int ra0 = (mq * 32 + l16) * KP;         // A frag row (M subtile 0 of this wave)
    const int ra1 = (mq * 32 + 16 + l16) * KP;    // A frag row (M subtile 1)
    const int rbn = nsub * 16 + l16;              // B frag col

    issue_chunk(0, 0);                        // prologue

    for (int ck = 0; ck < NCHUNK; ++ck) {
        const int cur = ck & 1;
        if (ck + 1 < NCHUNK) {
            issue_chunk(cur ^ 1, (ck + 1) * KC);         // safe: barrier at end of prev iter
            __builtin_amdgcn_s_wait_asynccnt(10);        // in-order => current chunk landed
        } else {
            __builtin_amdgcn_s_wait_asynccnt(0);
        }
        __syncthreads();                                  // all waves' async data visible

        #pragma unroll
        for (int kk = 0; kk < KC; kk += 4) {
            const int kb = kk + 2 * half;
            v2f a0, a1;
            a0.x = sA[cur][ra0 + kb];
            a0.y = sA[cur][ra0 + kb + 1];
            a1.x = sA[cur][ra1 + kb];
            a1.y = sA[cur][ra1 + kb + 1];
            #pragma unroll
            for (int g = 0; g < 4; ++g) {
                v2f b;
                b.x = sW[cur][(g * BN + rbn) * KP + kb];   // Whh[n][k] == Whh^T[k][n]
                b.y = sW[cur][(g * BN + rbn) * KP + kb + 1];
                acc[0][g] = __builtin_amdgcn_wmma_f32_16x16x4_f32(
                    false, a0, false, b, (short)0, acc[0][g], false, false);
                acc[1][g] = __builtin_amdgcn_wmma_f32_16x16x4_f32(
                    false, a1, false, b, (short)0, acc[1][g], false, false);
            }
        }
        __syncthreads();                                  // buffer may be refilled next iter
    }

    // epilogue: add x@Wih^T + biases, activations, c/h update
    const int j = n0 + nsub * 16 + l16;      // global hidden col for this lane
    #pragma unroll
    for (int mt = 0; mt < 2; ++mt) {
        float gv[4][8];
        #pragma unroll
        for (int g = 0; g < 4; ++g) {
            const float bsum = bih[g * HID + j] + bhh[g * HID + j];
            float wr[IN_DIM];
            #pragma unroll
            for (int k = 0; k < IN_DIM; ++k) wr[k] = Wih[(size_t)(g * HID + j) * IN_DIM + k];
            #pragma unroll
            for (int r = 0; r < 8; ++r) {
                const int m = mq * 32 + mt * 16 + half * 8 + r;   // C/D layout: M = r + 8*half
                float s = acc[mt][g][r] + bsum;
                #pragma unroll
                for (int k = 0; k < IN_DIM; ++k) s += sX[m * IN_DIM + k] * wr[k];
                gv[g][r] = s;
            }
        }
        #pragma unroll
        for (int r = 0; r < 8; ++r) {
            const int brow = b0 + mq * 32 + mt * 16 + half * 8 + r;
            const size_t idx = (size_t)brow * HID + j;
            const float ig = sigmoidf_(gv[0][r]);
            const float fg = sigmoidf_(gv[1][r]);
            const float gg = tanhf(gv[2][r]);
            const float og = sigmoidf_(gv[3][r]);
            const float cn = fg * c_state[idx] + ig * gg;
            c_state[idx] = cn;
            h_next[idx]  = og * tanhf(cn);
        }
    }
}

// out[:,t,:] = h @ Wlin^T + blin.  One wave per batch row. grid = BATCH/8, block = 256.
__global__ __launch_bounds__(256)
void out_proj_kernel(const float* __restrict__ h,     // [B,H]
                     const float* __restrict__ Wlin,  // [IN,H]
                     const float* __restrict__ blin,  // [IN]
                     float* __restrict__ out,         // [B,T,IN]
                     int t)
{
    const int wid  = threadIdx.x >> 5;
    const int lane = threadIdx.x & 31;
    const int b    = blockIdx.x * 8 + wid;
    const float* hr = h + (size_t)b * HID;

    float hv[32];
    #pragma unroll
    for (int i = 0; i < 32; ++i) hv[i] = hr[i * 32 + lane];

    for (int j = 0; j < IN_DIM; ++j) {
        const float* wr = Wlin + (size_t)j * HID;
        float s = 0.0f;
        #pragma unroll
        for (int i = 0; i < 32; ++i) s += hv[i] * wr[i * 32 + lane];
        #pragma unroll
        for (int off = 16; off > 0; off >>= 1) s += __shfl_xor(s, off, 32);
        if (lane == 0) out[((size_t)b * SEQ_T + t) * IN_DIM + j] = s + blin[j];
    }
}

__global__ void zero_kernel(float* __restrict__ p, size_t n) {
    size_t i = (size_t)blockIdx.x * blockDim.x + threadIdx.x;
    if (i < n) p[i] = 0.0f;
}

extern "C" void kernel_launch(void* const* d_in, const int* in_sizes, int n_in,
                              void* d_out, int out_size, void* d_ws, size_t ws_size,
                              hipStream_t stream) {
    const float* x    = (const float*)d_in[0];
    const float* Wih  = (const float*)d_in[1];
    const float* Whh  = (const float*)d_in[2];
    const float* bih  = (const float*)d_in[3];
    const float* bhh  = (const float*)d_in[4];
    const float* Wlin = (const float*)d_in[5];
    const float* blin = (const float*)d_in[6];
    float* out = (float*)d_out;

    const size_t NH = (size_t)BATCH * HID;
    // ws layout: [h_buf0][c][h_buf1]  (zero first two regions in one pass)
    float* h0 = (float*)d_ws;
    float* c  = h0 + NH;
    float* h1 = c + NH;
    float* bufs[2] = { h0, h1 };

    const size_t zn = 2 * NH;
    zero_kernel<<<(unsigned)((zn + 255) / 256), 256, 0, stream>>>(h0, zn);

    dim3 gridA(HID / BN, BATCH / BM);   // 16 x 64 = 1024 blocks
    for (int t = 0; t < SEQ_T; ++t) {
        const float* hp = bufs[t & 1];
        float* hn = bufs[(t + 1) & 1];
        lstm_step_kernel<<<gridA, 256, 0, stream>>>(x, Wih, Whh, bih, bhh, hp, hn, c, t);
        out_proj_kernel<<<BATCH / 8, 256, 0, stream>>>(hn, Wlin, blin, out, t);
    }

    // tuple output: [output (B*T*IN)] [h_fin (B*H)] [c_fin (B*H)]
    float* hfin = bufs[SEQ_T & 1];
    const size_t OUT0 = (size_t)BATCH * SEQ_T * IN_DIM;
    hipMemcpyAsync(out + OUT0,      hfin, NH * sizeof(float), hipMemcpyDeviceToDevice, stream);
    hipMemcpyAsync(out + OUT0 + NH, c,    NH * sizeof(float), hipMemcpyDeviceToDevice, stream);
}